// ZXNet_7627861918208
// MI455X (gfx1250) — compile-verified
//
#include <hip/hip_runtime.h>

typedef __attribute__((ext_vector_type(2))) float v2f;
typedef __attribute__((ext_vector_type(8))) float v8f;

#define NGRAPH 256  // G: graphs per batch side

// ---------------------------------------------------------------- utilities
__global__ void fill_f32(float* __restrict__ p, float v, int n) {
  int i = blockIdx.x * blockDim.x + threadIdx.x;
  if (i < n) p[i] = v;
}

__global__ void count_indeg(const int* __restrict__ dst, float* __restrict__ deg, int E) {
  int e = blockIdx.x * blockDim.x + threadIdx.x;
  if (e < E) atomicAdd(&deg[dst[e]], 1.0f);
}

__global__ void rsqrt_inplace(float* __restrict__ p, int n) {
  int i = blockIdx.x * blockDim.x + threadIdx.x;
  if (i < n) p[i] = rsqrtf(p[i]);
}

// ------------------------------------------------- node-feature GEMM (WMMA)
// C[n_tiles*16, 64] = op(A)[.,K] @ W[K,64]     op = relu if RELU_IN
// One wave handles one 16-row M-tile, looping over the 4 N-tiles of 64 cols.
// f32 WMMA 16x16x4: A tile per lane = v2f (lanes 0-15: K+{0,1}, lanes 16-31:
// K+{2,3}); B same K split; C/D v8f with M = r + 8*(lane>=16).
template<int K, bool RELU_IN>
__global__ void gemm_nodes(const float* __restrict__ A, const float* __restrict__ W,
                           float* __restrict__ C, int n_tiles)
{
  const int wave = blockIdx.x * (blockDim.x >> 5) + (threadIdx.x >> 5);
  const int lane = threadIdx.x & 31;
  if (wave >= n_tiles) return;               // uniform per wave -> EXEC all-1s below
  const int m  = lane & 15;
  const int hi = lane >> 4;                  // 0: lanes 0-15, 1: lanes 16-31
  const long row = (long)wave * 16 + m;
  constexpr int KS = K / 4;

  v2f a[KS];
#pragma unroll
  for (int k = 0; k < KS; ++k) {
    float x0 = A[row * K + 4 * k + 2 * hi + 0];
    float x1 = A[row * K + 4 * k + 2 * hi + 1];
    if (RELU_IN) { x0 = fmaxf(x0, 0.f); x1 = fmaxf(x1, 0.f); }
    v2f av; av.x = x0; av.y = x1;
    a[k] = av;
  }

#pragma unroll
  for (int nt = 0; nt < 4; ++nt) {
    v8f c = {};
#pragma unroll
    for (int k = 0; k < KS; ++k) {
      v2f b;
      b.x = W[(4 * k + 2 * hi + 0) * 64 + nt * 16 + m];
      b.y = W[(4 * k + 2 * hi + 1) * 64 + nt * 16 + m];
      c = __builtin_amdgcn_wmma_f32_16x16x4_f32(
              /*neg_a=*/false, a[k], /*neg_b=*/false, b,
              /*c_mod=*/(short)0, c, /*reuse_a=*/false, /*reuse_b=*/false);
    }
#pragma unroll
    for (int r = 0; r < 8; ++r)
      C[((long)wave * 16 + hi * 8 + r) * 64 + nt * 16 + m] = c[r];
  }
}

// ------------------------------------------- GCN aggregation (L2-resident)
// agg[n] = h[n]*dinv[n]^2 + bias   (self-loop term + bias)
__global__ void init_agg(const float* __restrict__ h, const float* __restrict__ dinv,
                         const float* __restrict__ bias, float* __restrict__ agg,
                         int n_elems /* = N*64 */)
{
  int i = blockIdx.x * blockDim.x + threadIdx.x;
  if (i >= n_elems) return;
  int nidx = i >> 6, f = i & 63;
  float di = dinv[nidx];
  agg[i] = h[i] * di * di + bias[f];
}

// agg[dst] += h[src] * dinv[src]*dinv[dst]; 16 threads per edge (float4 each)
__global__ void edge_scatter(const float* __restrict__ h, float* __restrict__ agg,
                             const int* __restrict__ src, const int* __restrict__ dst,
                             const float* __restrict__ dinv, int E)
{
  int idx = blockIdx.x * blockDim.x + threadIdx.x;
  int e = idx >> 4;
  if (e >= E) return;
  int q = idx & 15;
  int s = src[e], d = dst[e];
  float nrm = dinv[s] * dinv[d];
  float4 v = *((const float4*)(h + (long)s * 64) + q);
  float* ap = agg + (long)d * 64 + q * 4;
  atomicAdd(ap + 0, v.x * nrm);
  atomicAdd(ap + 1, v.y * nrm);
  atomicAdd(ap + 2, v.z * nrm);
  atomicAdd(ap + 3, v.w * nrm);
}

// ------------------------------------------------------- global mean pool
__global__ void pool_accum(const float* __restrict__ agg, const int* __restrict__ batch,
                           float* __restrict__ psum, float* __restrict__ pcnt, int n_nodes)
{
  int idx = blockIdx.x * blockDim.x + threadIdx.x;
  int nidx = idx >> 4;
  if (nidx >= n_nodes) return;
  int q = idx & 15;
  int g = batch[nidx];
  float4 v = *((const float4*)(agg + (long)nidx * 64) + q);
  float* p = psum + (long)g * 64 + q * 4;
  atomicAdd(p + 0, fmaxf(v.x, 0.f));   // relu fused into pooling
  atomicAdd(p + 1, fmaxf(v.y, 0.f));
  atomicAdd(p + 2, fmaxf(v.z, 0.f));
  atomicAdd(p + 3, fmaxf(v.w, 0.f));
  if (q == 0) atomicAdd(&pcnt[g], 1.0f);
}

// ----------------------------------------------------------- final MLP head
// out[g,:2] = relu([emb1|emb2] @ Wf1 + bf1) @ Wf2 + bf2
__global__ void final_mlp(const float* __restrict__ psum, const float* __restrict__ pcnt,
                          const float* __restrict__ Wf1, const float* __restrict__ bf1,
                          const float* __restrict__ Wf2, const float* __restrict__ bf2,
                          float* __restrict__ out)
{
  int g = blockIdx.x * blockDim.x + threadIdx.x;
  if (g >= NGRAPH) return;
  float emb[128];
  float c1 = fmaxf(pcnt[g], 1.f);
  float c2 = fmaxf(pcnt[NGRAPH + g], 1.f);
#pragma unroll
  for (int f = 0; f < 64; ++f) emb[f]      = psum[(long)g * 64 + f] / c1;
#pragma unroll
  for (int f = 0; f < 64; ++f) emb[64 + f] = psum[(long)(NGRAPH + g) * 64 + f] / c2;
  float o0 = bf2[0], o1 = bf2[1];
  for (int j = 0; j < 64; ++j) {
    float hsum = bf1[j];
#pragma unroll
    for (int i = 0; i < 128; ++i) hsum += emb[i] * Wf1[i * 64 + j];
    float hr = fmaxf(hsum, 0.f);
    o0 += hr * Wf2[j * 2 + 0];
    o1 += hr * Wf2[j * 2 + 1];
  }
  out[g * 2 + 0] = o0;
  out[g * 2 + 1] = o1;
}

// ------------------------------------------------------------------ driver
extern "C" void kernel_launch(void* const* d_in, const int* in_sizes, int n_in,
                              void* d_out, int out_size, void* d_ws, size_t ws_size,
                              hipStream_t stream)
{
  const int N = in_sizes[0] / 16;   // 100000
  const int E = in_sizes[1] / 2;    // 1600000

  const float* W1  = (const float*)d_in[6];
  const float* b1  = (const float*)d_in[7];
  const float* W2  = (const float*)d_in[8];
  const float* b2  = (const float*)d_in[9];
  const float* Wf1 = (const float*)d_in[10];
  const float* bf1 = (const float*)d_in[11];
  const float* Wf2 = (const float*)d_in[12];
  const float* bf2 = (const float*)d_in[13];

  // Workspace layout (floats): bufA[N*64] | bufB[N*64] | dinv[N] | psum[2*G*64] | pcnt[2*G]
  float* ws   = (float*)d_ws;
  float* bufA = ws;
  float* bufB = bufA + (size_t)N * 64;
  float* dinv = bufB + (size_t)N * 64;
  float* psum = dinv + N;
  float* pcnt = psum + 2 * NGRAPH * 64;

  const int TB = 256;
  // zero pooling accumulators (harness poisons ws with 0xAA)
  {
    int pn = 2 * NGRAPH * 64 + 2 * NGRAPH;
    fill_f32<<<(pn + TB - 1) / TB, TB, 0, stream>>>(psum, 0.f, pn);
  }

  const int ntiles  = N / 16;                 // 6250 (exact)
  const int gblocks = (ntiles + 7) / 8;       // 8 waves per 256-thread block

  for (int gi = 0; gi < 2; ++gi) {
    const float* x    = (const float*)d_in[gi * 3 + 0];
    const int*   ei   = (const int*)  d_in[gi * 3 + 1];
    const int*   bat  = (const int*)  d_in[gi * 3 + 2];
    const int*   srcp = ei;
    const int*   dstp = ei + E;

    // deg = 1 (self-loop) + in-degree; dinv = rsqrt(deg)
    fill_f32      <<<(N + TB - 1) / TB, TB, 0, stream>>>(dinv, 1.f, N);
    count_indeg   <<<(E + TB - 1) / TB, TB, 0, stream>>>(dstp, dinv, E);
    rsqrt_inplace <<<(N + TB - 1) / TB, TB, 0, stream>>>(dinv, N);

    // layer 1: h = x @ W1 ; agg = norm-aggregate(h) + b1
    gemm_nodes<16, false><<<gblocks, TB, 0, stream>>>(x, W1, bufA, ntiles);
    init_agg    <<<(N * 64 + TB - 1) / TB, TB, 0, stream>>>(bufA, dinv, b1, bufB, N * 64);
    edge_scatter<<<(E * 16 + TB - 1) / TB, TB, 0, stream>>>(bufA, bufB, srcp, dstp, dinv, E);

    // layer 2: h2 = relu(agg1) @ W2 ; agg2 = norm-aggregate(h2) + b2
    gemm_nodes<64, true><<<gblocks, TB, 0, stream>>>(bufB, W2, bufA, ntiles);
    init_agg    <<<(N * 64 + TB - 1) / TB, TB, 0, stream>>>(bufA, dinv, b2, bufB, N * 64);
    edge_scatter<<<(E * 16 + TB - 1) / TB, TB, 0, stream>>>(bufA, bufB, srcp, dstp, dinv, E);

    // global mean pool of relu(agg2)
    pool_accum<<<(N * 16 + TB - 1) / TB, TB, 0, stream>>>(
        bufB, bat, psum + (size_t)gi * NGRAPH * 64, pcnt + (size_t)gi * NGRAPH, N);
  }

  final_mlp<<<1, TB, 0, stream>>>(psum, pcnt, Wf1, bf1, Wf2, bf2, (float*)d_out);
}